// ChamferDistanceL2_49177375539582
// MI455X (gfx1250) — compile-verified
//
#include <hip/hip_runtime.h>

typedef __attribute__((ext_vector_type(2))) float v2f;
typedef __attribute__((ext_vector_type(8))) float v8f;

#define B_ 8
#define N_ 8192
#define M_ 8192

// Pack each 3-D point p into float4: (-2px, -2py, -2pz, ||p||^2).
//  - Used as B-operand directly (columns [-2y0,-2y1,-2y2,||y||^2]).
//  - A-operand recovered as (-0.5*x, -0.5*y) / (-0.5*z, 1.0) per lane half.
__global__ __launch_bounds__(256) void chamfer_pack(const float* __restrict__ xyz,
                                                    float4* __restrict__ P, int total) {
    int i = blockIdx.x * 256 + threadIdx.x;
    if (i >= total) return;
    float x = xyz[3 * i + 0];
    float y = xyz[3 * i + 1];
    float z = xyz[3 * i + 2];
    P[i] = make_float4(-2.0f * x, -2.0f * y, -2.0f * z, x * x + y * y + z * z);
}

// For each point in set A, min squared distance to all points in set B.
// One wave = 16 A-rows; loops over all B points in 16-wide tiles via
// V_WMMA_F32_16X16X4_F32 (K=4: 3 coords + norm trick -> full distance tile).
__global__ __launch_bounds__(256) void chamfer_nn(const float4* __restrict__ PA,
                                                  const float4* __restrict__ PB,
                                                  float* __restrict__ dist,
                                                  int nA, int nB) {
    const int lane = threadIdx.x & 31;
    const int wave = threadIdx.x >> 5;
    const int half = lane >> 4;   // 0: lanes 0-15 (K=0,1), 1: lanes 16-31 (K=2,3)
    const int l16  = lane & 15;
    const int b    = blockIdx.y;
    const int rowbase = blockIdx.x * 128 + wave * 16;

    const float4* pa = PA + (size_t)b * nA;
    const float4* pb = PB + (size_t)b * nB;

    // A-operand: row (M) = rowbase + l16 for both lane halves.
    // lanes 0-15 hold (x0, x1); lanes 16-31 hold (x2, 1). Branchless (cndmask,
    // no EXEC change) so EXEC is all-1s at the WMMAs.
    float4 qa = pa[rowbase + l16];
    v2f a;
    a[0] = half ? (-0.5f * qa.z) : (-0.5f * qa.x);
    a[1] = half ? 1.0f           : (-0.5f * qa.y);

    v8f rmin;
#pragma unroll
    for (int i = 0; i < 8; ++i) rmin[i] = 3.4e38f;

    const float* pbf = (const float*)pb;
    // Per tile: lane reads float2 of PB[mt + l16]: components (0,1) or (2,3).
    // Wave reads 256 contiguous bytes -> one coalesced b64 load per lane.
#pragma unroll 4
    for (int mt = 0; mt < nB; mt += 16) {
        const float* src = pbf + (size_t)(mt + l16) * 4 + half * 2;
        v2f bv = *(const v2f*)src;
        v8f c = {};
        // D = A(16x4) * B(4x16) + 0 = -2 x.y + ||y||^2 per 16x16 tile
        v8f d = __builtin_amdgcn_wmma_f32_16x16x4_f32(
            /*neg_a=*/false, a, /*neg_b=*/false, bv,
            /*c_mod=*/(short)0, c, /*reuse_a=*/false, /*reuse_b=*/false);
#pragma unroll
        for (int i = 0; i < 8; ++i) rmin[i] = fminf(rmin[i], d[i]);
    }

    // Min-reduce the 16 columns within each 16-lane half (xor 1,2,4,8 stays
    // inside the half). C/D layout: VGPR v, lanes 0-15 -> M=v; lanes 16-31 -> M=v+8.
#pragma unroll
    for (int off = 1; off < 16; off <<= 1) {
#pragma unroll
        for (int i = 0; i < 8; ++i)
            rmin[i] = fminf(rmin[i], __shfl_xor(rmin[i], off, 32));
    }

    if (l16 == 0) {  // lanes 0 and 16 write 8 rows each (static unroll, no
                     // dynamic register indexing)
#pragma unroll
        for (int v = 0; v < 8; ++v) {
            int row = rowbase + v + 8 * half;
            dist[(size_t)b * nA + row] = rmin[v] + pa[row].w;  // + ||x||^2
        }
    }
}

// Deterministic final reduction: fixed-stride accumulation + LDS tree.
// dist1 and dist2 are contiguous; N==M so result = (sum1+sum2)/(B*N).
__global__ __launch_bounds__(256) void chamfer_reduce(const float* __restrict__ dists,
                                                      int n, float scale,
                                                      float* __restrict__ out) {
    __shared__ float s[256];
    float acc = 0.0f;
    for (int i = threadIdx.x; i < n; i += 256) acc += dists[i];
    s[threadIdx.x] = acc;
    __syncthreads();
    for (int o = 128; o > 0; o >>= 1) {
        if ((int)threadIdx.x < o) s[threadIdx.x] += s[threadIdx.x + o];
        __syncthreads();
    }
    if (threadIdx.x == 0) out[0] = s[0] * scale;
}

extern "C" void kernel_launch(void* const* d_in, const int* in_sizes, int n_in,
                              void* d_out, int out_size, void* d_ws, size_t ws_size,
                              hipStream_t stream) {
    const float* xyz1 = (const float*)d_in[0];  // [B, N, 3] fp32
    const float* xyz2 = (const float*)d_in[1];  // [B, M, 3] fp32
    float* out = (float*)d_out;                 // scalar fp32

    char* ws = (char*)d_ws;
    const size_t pkBytes = (size_t)B_ * N_ * sizeof(float4);  // 1 MB each
    float4* P1 = (float4*)(ws);
    float4* P2 = (float4*)(ws + pkBytes);
    float* dist1 = (float*)(ws + 2 * pkBytes);        // [B*N]
    float* dist2 = dist1 + (size_t)B_ * N_;           // [B*M], contiguous

    const int total = B_ * N_;
    chamfer_pack<<<dim3((total + 255) / 256), dim3(256), 0, stream>>>(xyz1, P1, total);
    chamfer_pack<<<dim3((total + 255) / 256), dim3(256), 0, stream>>>(xyz2, P2, total);

    dim3 grid(N_ / 128, B_);  // 8 waves/block, 16 rows/wave
    chamfer_nn<<<grid, dim3(256), 0, stream>>>(P1, P2, dist1, N_, M_);
    chamfer_nn<<<grid, dim3(256), 0, stream>>>(P2, P1, dist2, M_, N_);

    chamfer_reduce<<<dim3(1), dim3(256), 0, stream>>>(dist1, 2 * B_ * N_,
                                                      1.0f / (float)(B_ * N_), out);
}